// MUTAG_GNN_5540507812347
// MI455X (gfx1250) — compile-verified
//
#include <hip/hip_runtime.h>
#include <hip/hip_bf16.h>

// ---------------------------------------------------------------------------
// MUTAG-style GNN: 5x GraphConv (scatter-add + dual GEMM + bias + ReLU),
// sum-pool per graph, lin1+ReLU (WMMA), lin2 + log_softmax.
// Dense parts: V_WMMA_F32_16X16X4_F32, fully unrolled, no predication.
// ---------------------------------------------------------------------------

#define NUM_NODES  100000
#define NUM_EDGES  2500000
#define NUM_GRAPHS 512
#define IN_DIM     14
#define INP        16      // IN_DIM zero-padded to WMMA-friendly multiple of 4
#define HID        32

typedef float v2f __attribute__((ext_vector_type(2)));
typedef float v8f __attribute__((ext_vector_type(8)));

__device__ __forceinline__ v8f wmma4(v2f a, v2f b, v8f c) {
    return __builtin_amdgcn_wmma_f32_16x16x4_f32(false, a, false, b,
                                                 (short)0, c, false, false);
}

// ---------------------------------------------------------------------------
// Pad x [N x 14] -> xpad [N x 16] (cols 14,15 = 0)
// ---------------------------------------------------------------------------
__global__ void pad_x_kernel(const float* __restrict__ x,
                             float* __restrict__ xpad) {
    int idx = blockIdx.x * blockDim.x + threadIdx.x;
    int node = idx >> 4;
    int f    = idx & 15;
    if (node < NUM_NODES)
        xpad[node * INP + f] = (f < IN_DIM) ? x[node * IN_DIM + f] : 0.0f;
}

// Pad W [14 x 32] -> Wp [16 x 32] (rows 14,15 = 0)
__global__ void pad_w_kernel(const float* __restrict__ w,
                             float* __restrict__ wp) {
    int idx = blockIdx.x * blockDim.x + threadIdx.x;  // 512 threads
    int k = idx >> 5;
    int n = idx & 31;
    if (k < INP)
        wp[k * HID + n] = (k < IN_DIM) ? w[k * HID + n] : 0.0f;
}

// ---------------------------------------------------------------------------
// Edge scatter-add, F = feature width (16 or 32, power of two).
// F lanes per edge; coalesced gather of h[src], float atomics into agg[dst].
// ---------------------------------------------------------------------------
template<int F>
__global__ void scatter_add_kernel(const float* __restrict__ h,
                                   const long long* __restrict__ src,
                                   const long long* __restrict__ dst,
                                   float* __restrict__ agg) {
    long long idx = (long long)blockIdx.x * blockDim.x + threadIdx.x;
    long long e = idx / F;
    int f       = (int)(idx % F);
    if (e < NUM_EDGES) {
        long long s = src[e];
        long long d = dst[e];
        atomicAdd(&agg[d * F + f], h[s * F + f]);
    }
}

// ---------------------------------------------------------------------------
// Sum-pool nodes into graphs: lane = feature, one wave per node.
// ---------------------------------------------------------------------------
__global__ void pool_add_kernel(const float* __restrict__ h,
                                const long long* __restrict__ batch,
                                float* __restrict__ g) {
    long long idx = (long long)blockIdx.x * blockDim.x + threadIdx.x;
    int node = (int)(idx >> 5);
    int f    = (int)(idx & 31);
    if (node < NUM_NODES) {
        long long b = batch[node];
        atomicAdd(&g[b * HID + f], h[node * HID + f]);
    }
}

// ---------------------------------------------------------------------------
// WMMA GEMM: out[M x 32] = act( A1[M x K1] @ B1[K1 x 32]
//                             (+ A2[M x K2] @ B2[K2 x 32]) + bias )
// K1/K2 compile-time multiples of 4 -> fully unrolled, no predication.
// One wave owns TWO 16-row tiles (32 rows x 32 cols); M % 32 == 0.
//
// Layouts (CDNA5 ISA 7.12.2, 32-bit, wave32):
//   A (16x4): lane L holds row m = L%16; vgpr j holds k = 2*(L/16) + j
//   B (4x16): lane L holds col n = L%16; vgpr j holds k = 2*(L/16) + j
//   C/D (16x16): lane L col n = L%16; vgpr r holds row m = r + 8*(L/16)
// ---------------------------------------------------------------------------
template<int K1, int K2>
__global__ __launch_bounds__(256)
void wmma_gemm_kernel(const float* __restrict__ A1, int ldA1,
                      const float* __restrict__ B1,
                      const float* __restrict__ A2, int ldA2,
                      const float* __restrict__ B2,
                      const float* __restrict__ bias,
                      float* __restrict__ out, int M, int do_relu) {
    int wave = (int)((blockIdx.x * blockDim.x + threadIdx.x) >> 5);
    int lane = threadIdx.x & 31;
    int row0 = wave * 32;                  // two 16-row tiles per wave
    if (row0 >= M) return;                 // wave-uniform: EXEC stays full

    int mr = lane & 15;                    // A row within tile
    int kh = (lane >> 4) << 1;             // lane-half K offset (0 or 2)
    int n  = lane & 15;                    // B/C column within half

    v8f acc[2][2];                         // [row tile][col half]
    acc[0][0] = {}; acc[0][1] = {}; acc[1][0] = {}; acc[1][1] = {};

    const float* a1p0 = A1 + (long long)(row0 + mr)      * ldA1;
    const float* a1p1 = A1 + (long long)(row0 + 16 + mr) * ldA1;

    #pragma unroll
    for (int kk = 0; kk < K1; kk += 4) {
        int k0 = kk + kh;
        v2f a0 = *(const v2f*)(a1p0 + k0);
        v2f a1 = *(const v2f*)(a1p1 + k0);
        v2f bLo, bHi;
        bLo.x = B1[k0 * HID + n];
        bLo.y = B1[(k0 + 1) * HID + n];
        bHi.x = B1[k0 * HID + 16 + n];
        bHi.y = B1[(k0 + 1) * HID + 16 + n];
        acc[0][0] = wmma4(a0, bLo, acc[0][0]);
        acc[0][1] = wmma4(a0, bHi, acc[0][1]);
        acc[1][0] = wmma4(a1, bLo, acc[1][0]);
        acc[1][1] = wmma4(a1, bHi, acc[1][1]);
    }

    if constexpr (K2 > 0) {
        const float* a2p0 = A2 + (long long)(row0 + mr)      * ldA2;
        const float* a2p1 = A2 + (long long)(row0 + 16 + mr) * ldA2;
        #pragma unroll
        for (int kk = 0; kk < K2; kk += 4) {
            int k0 = kk + kh;
            v2f a0 = *(const v2f*)(a2p0 + k0);
            v2f a1 = *(const v2f*)(a2p1 + k0);
            v2f bLo, bHi;
            bLo.x = B2[k0 * HID + n];
            bLo.y = B2[(k0 + 1) * HID + n];
            bHi.x = B2[k0 * HID + 16 + n];
            bHi.y = B2[(k0 + 1) * HID + 16 + n];
            acc[0][0] = wmma4(a0, bLo, acc[0][0]);
            acc[0][1] = wmma4(a0, bHi, acc[0][1]);
            acc[1][0] = wmma4(a1, bLo, acc[1][0]);
            acc[1][1] = wmma4(a1, bHi, acc[1][1]);
        }
    }

    // --- epilogue: bias, ReLU, store (C/D layout; M % 32 == 0 so no guards)
    int half = lane >> 4;
    float bL = bias[n];
    float bH = bias[16 + n];
    #pragma unroll
    for (int t = 0; t < 2; ++t) {
        #pragma unroll
        for (int r = 0; r < 8; ++r) {
            int mrow = row0 + t * 16 + r + half * 8;
            float vL = acc[t][0][r] + bL;
            float vH = acc[t][1][r] + bH;
            if (do_relu) { vL = fmaxf(vL, 0.0f); vH = fmaxf(vH, 0.0f); }
            out[(long long)mrow * HID + n]      = vL;
            out[(long long)mrow * HID + 16 + n] = vH;
        }
    }
}

// ---------------------------------------------------------------------------
// Head: logits = g @ lin2_w[32x2] + lin2_b; log_softmax over 2 classes.
// ---------------------------------------------------------------------------
__global__ void head_kernel(const float* __restrict__ g,
                            const float* __restrict__ w2,
                            const float* __restrict__ b2,
                            float* __restrict__ out, int ngraphs) {
    int i = blockIdx.x * blockDim.x + threadIdx.x;
    if (i >= ngraphs) return;
    float l0 = b2[0];
    float l1 = b2[1];
    #pragma unroll
    for (int k = 0; k < HID; ++k) {
        float v = g[i * HID + k];
        l0 += v * w2[k * 2 + 0];
        l1 += v * w2[k * 2 + 1];
    }
    float mx  = fmaxf(l0, l1);
    float lse = mx + __logf(__expf(l0 - mx) + __expf(l1 - mx));
    out[i * 2 + 0] = l0 - lse;
    out[i * 2 + 1] = l1 - lse;
}

// ---------------------------------------------------------------------------
// Host orchestration
// ---------------------------------------------------------------------------
extern "C" void kernel_launch(void* const* d_in, const int* in_sizes, int n_in,
                              void* d_out, int out_size, void* d_ws, size_t ws_size,
                              hipStream_t stream) {
    (void)in_sizes; (void)n_in; (void)out_size; (void)ws_size;

    const float*     x       = (const float*)d_in[0];      // [100000,14]
    const float*     W_rel1  = (const float*)d_in[1];      // [14,32]
    const float*     b_rel1  = (const float*)d_in[2];      // [32]
    const float*     W_root1 = (const float*)d_in[3];      // [14,32]
    const float*     W_rel   = (const float*)d_in[4];      // [4,32,32]
    const float*     b_rel   = (const float*)d_in[5];      // [4,32]
    const float*     W_root  = (const float*)d_in[6];      // [4,32,32]
    const float*     lin1_w  = (const float*)d_in[7];      // [32,32]
    const float*     lin1_b  = (const float*)d_in[8];      // [32]
    const float*     lin2_w  = (const float*)d_in[9];      // [32,2]
    const float*     lin2_b  = (const float*)d_in[10];     // [2]
    const long long* eidx    = (const long long*)d_in[11]; // [2, 2.5M] int64
    const long long* batch   = (const long long*)d_in[12]; // [100000] int64

    const long long* src = eidx;
    const long long* dst = eidx + NUM_EDGES;

    // workspace carve-out (all offsets well aligned)
    char* ws = (char*)d_ws;
    const size_t H_BYTES  = (size_t)NUM_NODES * HID * sizeof(float); // 12.8 MB
    const size_t XP_BYTES = (size_t)NUM_NODES * INP * sizeof(float); //  6.4 MB
    float* hA      = (float*)(ws);
    float* hB      = (float*)(ws + H_BYTES);
    float* agg     = (float*)(ws + 2 * H_BYTES);  // reused as agg16 (smaller)
    float* xpad    = (float*)(ws + 3 * H_BYTES);
    float* Wrel1p  = (float*)(ws + 3 * H_BYTES + XP_BYTES);
    float* Wroot1p = (float*)(ws + 3 * H_BYTES + XP_BYTES + 4096);
    float* g       = (float*)(ws + 3 * H_BYTES + XP_BYTES + 8192);
    float* gLin    = (float*)(ws + 3 * H_BYTES + XP_BYTES + 8192 + 65536);

    const int BLK = 256;
    const int SC16_GRID = (int)(((long long)NUM_EDGES * INP + BLK - 1) / BLK);
    const int SC32_GRID = (int)(((long long)NUM_EDGES * HID + BLK - 1) / BLK);
    const int PL_GRID   = (int)(((long long)NUM_NODES * HID + BLK - 1) / BLK);
    const int PX_GRID   = (NUM_NODES * INP + BLK - 1) / BLK;

    // GEMM: 8 waves/block, 32 rows/wave -> 256 rows/block
    const int GM_GRID   = (NUM_NODES / 32 + 7) / 8;   // 391
    const int GM_GRID_G = (NUM_GRAPHS / 32 + 7) / 8;  // 2

    // ---- pad x and layer-1 weights to K=16
    pad_x_kernel<<<PX_GRID, BLK, 0, stream>>>(x, xpad);
    pad_w_kernel<<<2, BLK, 0, stream>>>(W_rel1, Wrel1p);
    pad_w_kernel<<<2, BLK, 0, stream>>>(W_root1, Wroot1p);

    // ---- layer 1: agg16 = scatter(xpad); h = relu(agg16@Wrel1p + b + xpad@Wroot1p)
    hipMemsetAsync(agg, 0, XP_BYTES, stream);
    scatter_add_kernel<INP><<<SC16_GRID, BLK, 0, stream>>>(xpad, src, dst, agg);
    wmma_gemm_kernel<INP, INP><<<GM_GRID, BLK, 0, stream>>>(
        agg, INP, Wrel1p, xpad, INP, Wroot1p, b_rel1, hA, NUM_NODES, 1);

    // ---- layers 2..5: ping-pong hA <-> hB
    float* cur = hA;
    float* nxt = hB;
    for (int i = 0; i < 4; ++i) {
        hipMemsetAsync(agg, 0, H_BYTES, stream);
        scatter_add_kernel<HID><<<SC32_GRID, BLK, 0, stream>>>(cur, src, dst, agg);
        wmma_gemm_kernel<HID, HID><<<GM_GRID, BLK, 0, stream>>>(
            agg, HID, W_rel + i * HID * HID,
            cur, HID, W_root + i * HID * HID,
            b_rel + i * HID, nxt, NUM_NODES, 1);
        float* t = cur; cur = nxt; nxt = t;
    }
    // after 4 swaps, cur == hA holds the final node features

    // ---- sum pool per graph
    hipMemsetAsync(g, 0, (size_t)NUM_GRAPHS * HID * sizeof(float), stream);
    pool_add_kernel<<<PL_GRID, BLK, 0, stream>>>(cur, batch, g);

    // ---- lin1 + ReLU (WMMA)
    wmma_gemm_kernel<HID, 0><<<GM_GRID_G, BLK, 0, stream>>>(
        g, HID, lin1_w, nullptr, 0, nullptr, lin1_b, gLin, NUM_GRAPHS, 1);

    // ---- lin2 + log_softmax
    head_kernel<<<(NUM_GRAPHS + BLK - 1) / BLK, BLK, 0, stream>>>(
        gLin, lin2_w, lin2_b, (float*)d_out, NUM_GRAPHS);
}